// ReceptorEncoder_64982855188923
// MI455X (gfx1250) — compile-verified
//
#include <hip/hip_runtime.h>
#include <hip/hip_bf16.h>
#include <math.h>

// ---------------------------------------------------------------------------
// ReceptorEncoder fused pipeline for gfx1250 (MI455X), wave32.
// Main GEMM uses V_WMMA_F32_16X16X4_F32 (exact fp32 numerics, HBM-bound).
// ---------------------------------------------------------------------------

typedef __attribute__((ext_vector_type(2))) float v2f;
typedef __attribute__((ext_vector_type(8))) float v8f;

#define BATCH   8
#define NREC    4096
#define NKP     20
#define FDIM    256
#define NHEADS  4
#define NCOL    (NKP * NHEADS)   // 80 score columns = (k,h)
#define KCL     8

// ws layout (floats)
#define WS_CT   0                          // C_T: B*80*256 = 163840
#define WS_S    163840                     // S:   B*80*4   = 2560
#define WS_DK   166400                     // d_k: B*20*8   = 1280
#define WS_IDX  167680                     // idx: B*20*8 ints

// ---------------------------------------------------------------------------
// Kernel 1: per (b,k,h) compute ft_dst row, then collapsed score matrix
//   C_T[b][k*4+h][f] = sum_d W_src[f][h*256+d] * ft_dst[b,k,h*256+d]
// Also zero-init the softmax accumulators S (2560 floats) via blocks 0..9.
// ---------------------------------------------------------------------------
__global__ __launch_bounds__(256) void prep_kernel(
    const float* __restrict__ h0_kp, const float* __restrict__ W_src,
    float* __restrict__ C_T, float* __restrict__ S)
{
    __shared__ float xk[FDIM];    // h0_kp row
    __shared__ float ftd[FDIM];   // ft_dst slice for this head

    const int bid = blockIdx.x;         // 0..639
    const int t   = threadIdx.x;        // 0..255

    if (bid < 10) S[bid * 256 + t] = 0.0f;   // 10*256 == 2560 exactly

    const int b   = bid / NCOL;
    const int col = bid % NCOL;         // col = k*4 + h
    const int k   = col >> 2;
    const int h   = col & 3;

    xk[t] = h0_kp[((size_t)b * NKP + k) * FDIM + t];
    __syncthreads();

    // ft_dst[d=t] = sum_f xk[f] * W_src[f][h*256 + t]   (W_src row-major 256x1024)
    {
        float acc = 0.0f;
        const float* Wc = W_src + h * FDIM + t;
        #pragma unroll 4
        for (int f = 0; f < FDIM; ++f) acc += xk[f] * Wc[(size_t)f * (FDIM * NHEADS)];
        ftd[t] = acc;
    }
    __syncthreads();

    // C_T[b][col][f=t] = sum_d W_src[t][h*256+d] * ftd[d]
    {
        float acc = 0.0f;
        const float* Wr = W_src + (size_t)t * (FDIM * NHEADS) + h * FDIM;
        #pragma unroll 4
        for (int d = 0; d < FDIM; ++d) acc += Wr[d] * ftd[d];
        C_T[((size_t)b * NCOL + col) * FDIM + t] = acc;
    }
}

// ---------------------------------------------------------------------------
// Kernel 2: fused score GEMM + exp + segment-softmax accumulation.
// Per batch: scores = h_rec[4096x256] @ C_T^T[256x80], via f32 WMMA 16x16x4.
// grid = B*32 blocks of 256 threads (8 waves); each wave owns a 16-row strip
// and all 5 column tiles (5 x v8f accumulators).
// Epilogue: w = exp(score/16); accumulate S0 += w, S1c += w * x_rec[r][c]
// per (b,col) with a cross-half shuffle reduction + global float atomics.
// ---------------------------------------------------------------------------
__global__ __launch_bounds__(256) void score_kernel(
    const float* __restrict__ h_rec, const float* __restrict__ x_rec,
    const float* __restrict__ C_T, float* __restrict__ S)
{
    const int b      = blockIdx.x >> 5;          // 0..7
    const int rt8    = blockIdx.x & 31;
    const int wave   = threadIdx.x >> 5;         // 0..7
    const int lane   = threadIdx.x & 31;
    const int lane16 = lane & 15;
    const int half   = lane >> 4;                // 0 or 1
    const int row0   = (rt8 * 8 + wave) * 16;    // 0..4080
    const int kk     = 2 * half;                 // K sub-offset for this half

    const float* __restrict__ Arow = h_rec + ((size_t)b * NREC + row0 + lane16) * FDIM;
    const float* __restrict__ Cb   = C_T + (size_t)b * NCOL * FDIM;

    v8f acc[5];
    #pragma unroll
    for (int ct = 0; ct < 5; ++ct) acc[ct] = (v8f){0.f,0.f,0.f,0.f,0.f,0.f,0.f,0.f};

    for (int k0 = 0; k0 < FDIM; k0 += 4) {
        // A fragment: A[m=lane16][k0+kk .. +1]
        v2f a = *(const v2f*)(Arow + k0 + kk);
        #pragma unroll
        for (int ct = 0; ct < 5; ++ct) {
            // B fragment: B[k][n=lane16] = C_T[b][ct*16+n][k0+kk .. +1]
            v2f bf = *(const v2f*)(Cb + (size_t)(ct * 16 + lane16) * FDIM + k0 + kk);
            acc[ct] = __builtin_amdgcn_wmma_f32_16x16x4_f32(
                false, a, false, bf, (short)0, acc[ct], false, false);
        }
    }

    // x_rec rows covered by this lane (m = v + 8*half)
    float xw[8][3];
    #pragma unroll
    for (int v = 0; v < 8; ++v) {
        const float* xp = x_rec + ((size_t)b * NREC + row0 + v + 8 * half) * 3;
        xw[v][0] = xp[0]; xw[v][1] = xp[1]; xw[v][2] = xp[2];
    }

    const float inv_sqrtD = 0.0625f;  // 1/sqrt(256)
    #pragma unroll
    for (int ct = 0; ct < 5; ++ct) {
        float s0 = 0.f, sx = 0.f, sy = 0.f, sz = 0.f;
        #pragma unroll
        for (int v = 0; v < 8; ++v) {
            float w = __expf(acc[ct][v] * inv_sqrtD);
            s0 += w;
            sx += w * xw[v][0];
            sy += w * xw[v][1];
            sz += w * xw[v][2];
        }
        // combine the two 8-row halves (lane L <-> L^16 share the column)
        s0 += __shfl_xor(s0, 16, 32);
        sx += __shfl_xor(sx, 16, 32);
        sy += __shfl_xor(sy, 16, 32);
        sz += __shfl_xor(sz, 16, 32);
        if (half == 0) {
            float* Sp = S + ((size_t)b * NCOL + ct * 16 + lane16) * 4;
            atomicAdd(Sp + 0, s0);
            atomicAdd(Sp + 1, sx);
            atomicAdd(Sp + 2, sy);
            atomicAdd(Sp + 3, sz);
        }
    }
}

// ---------------------------------------------------------------------------
// Kernel 3: kp_pos[b,k,c] = mean_h ( S1c[b,k,h] / S0[b,k,h] )
// ---------------------------------------------------------------------------
__global__ __launch_bounds__(256) void kppos_kernel(
    const float* __restrict__ S, float* __restrict__ kp_pos)
{
    int t = blockIdx.x * 256 + threadIdx.x;
    if (t >= BATCH * NKP * 3) return;
    int b = t / (NKP * 3);
    int rem = t % (NKP * 3);
    int k = rem / 3, c = rem % 3;
    float acc = 0.f;
    #pragma unroll
    for (int h = 0; h < NHEADS; ++h) {
        const float* Sp = S + ((size_t)b * NCOL + k * NHEADS + h) * 4;
        acc += Sp[1 + c] / Sp[0];
    }
    kp_pos[((size_t)b * NKP + k) * 3 + c] = acc * 0.25f;
}

// ---------------------------------------------------------------------------
// Kernel 4: top-8 closest rec atoms per keypoint. One block per (b,k).
// Per-thread sorted top-8 over a strided slice, then LDS merge-tree.
// Stable tie-break on lower index (matches jax.lax.top_k).
// ---------------------------------------------------------------------------
__global__ __launch_bounds__(256) void topk_kernel(
    const float* __restrict__ x_rec, const float* __restrict__ kp_pos,
    float* __restrict__ dk_out, int* __restrict__ idx_out)
{
    __shared__ float sd[256 * KCL];
    __shared__ int   si[256 * KCL];

    const int bk = blockIdx.x;         // 0..159
    const int b  = bk / NKP;
    const int t  = threadIdx.x;

    const float px = kp_pos[bk * 3 + 0];
    const float py = kp_pos[bk * 3 + 1];
    const float pz = kp_pos[bk * 3 + 2];

    float d[KCL]; int id[KCL];
    #pragma unroll
    for (int j = 0; j < KCL; ++j) { d[j] = 3.4e38f; id[j] = 0x7fffffff; }

    const float* xb = x_rec + (size_t)b * NREC * 3;
    for (int r = t; r < NREC; r += 256) {
        float dx = xb[r * 3 + 0] - px + 1e-30f;
        float dy = xb[r * 3 + 1] - py + 1e-30f;
        float dz = xb[r * 3 + 2] - pz + 1e-30f;
        float dist = sqrtf(dx * dx + dy * dy + dz * dz);
        if (dist < d[KCL - 1]) {
            int pos = KCL - 1;
            while (pos > 0 && dist < d[pos - 1]) {
                d[pos] = d[pos - 1]; id[pos] = id[pos - 1]; --pos;
            }
            d[pos] = dist; id[pos] = r;
        }
    }
    #pragma unroll
    for (int j = 0; j < KCL; ++j) { sd[t * KCL + j] = d[j]; si[t * KCL + j] = id[j]; }
    __syncthreads();

    for (int stride = 128; stride > 0; stride >>= 1) {
        if (t < stride) {
            float da[KCL], db[KCL], dm[KCL];
            int   ia[KCL], ib[KCL], im[KCL];
            #pragma unroll
            for (int j = 0; j < KCL; ++j) {
                da[j] = sd[t * KCL + j];            ia[j] = si[t * KCL + j];
                db[j] = sd[(t + stride) * KCL + j]; ib[j] = si[(t + stride) * KCL + j];
            }
            int pa = 0, pb = 0;
            #pragma unroll
            for (int j = 0; j < KCL; ++j) {
                bool takeA = (pb >= KCL) ||
                             ((pa < KCL) && ((da[pa] < db[pb]) ||
                                             (da[pa] == db[pb] && ia[pa] < ib[pb])));
                if (takeA) { dm[j] = da[pa]; im[j] = ia[pa]; ++pa; }
                else       { dm[j] = db[pb]; im[j] = ib[pb]; ++pb; }
            }
            #pragma unroll
            for (int j = 0; j < KCL; ++j) { sd[t * KCL + j] = dm[j]; si[t * KCL + j] = im[j]; }
        }
        __syncthreads();
    }
    if (t < KCL) {
        dk_out[bk * KCL + t]  = sd[t];
        idx_out[bk * KCL + t] = si[t];
    }
}

// ---------------------------------------------------------------------------
// Kernel 5: gather-mean of 8 rec rows, 264->256 MLP + SiLU + LayerNorm.
// One block per (b,k); thread j owns output feature j.
// ---------------------------------------------------------------------------
__global__ __launch_bounds__(256) void mlp_kernel(
    const float* __restrict__ h_rec, const float* __restrict__ W_mlp,
    const float* __restrict__ b_mlp, const float* __restrict__ ln_g,
    const float* __restrict__ ln_b, const float* __restrict__ dk,
    const int* __restrict__ idx, float* __restrict__ kp_feat)
{
    __shared__ float feat[FDIM + KCL];
    __shared__ float red[256];

    const int bk = blockIdx.x;
    const int b  = bk / NKP;
    const int t  = threadIdx.x;

    const float* hb = h_rec + (size_t)b * NREC * FDIM;
    float s = 0.f;
    #pragma unroll
    for (int j = 0; j < KCL; ++j) s += hb[(size_t)idx[bk * KCL + j] * FDIM + t];
    feat[t] = s * (1.0f / KCL);
    if (t < KCL) feat[FDIM + t] = dk[bk * KCL + t];
    __syncthreads();

    float o = b_mlp[t];
    #pragma unroll 4
    for (int i = 0; i < FDIM + KCL; ++i) o += feat[i] * W_mlp[(size_t)i * FDIM + t];
    o = o / (1.0f + __expf(-o));               // SiLU

    // LayerNorm (biased variance, eps=1e-5)
    red[t] = o; __syncthreads();
    for (int st = 128; st > 0; st >>= 1) { if (t < st) red[t] += red[t + st]; __syncthreads(); }
    float mean = red[0] * (1.0f / FDIM);
    __syncthreads();
    float c = o - mean;
    red[t] = c * c; __syncthreads();
    for (int st = 128; st > 0; st >>= 1) { if (t < st) red[t] += red[t + st]; __syncthreads(); }
    float var = red[0] * (1.0f / FDIM);

    kp_feat[(size_t)bk * FDIM + t] = c * rsqrtf(var + 1e-5f) * ln_g[t] + ln_b[t];
}

// ---------------------------------------------------------------------------
extern "C" void kernel_launch(void* const* d_in, const int* in_sizes, int n_in,
                              void* d_out, int out_size, void* d_ws, size_t ws_size,
                              hipStream_t stream) {
    const float* h_rec = (const float*)d_in[0];
    const float* x_rec = (const float*)d_in[1];
    const float* h0_kp = (const float*)d_in[2];
    const float* W_src = (const float*)d_in[3];
    const float* W_mlp = (const float*)d_in[4];
    const float* b_mlp = (const float*)d_in[5];
    const float* ln_g  = (const float*)d_in[6];
    const float* ln_b  = (const float*)d_in[7];
    (void)in_sizes; (void)n_in; (void)out_size; (void)ws_size;

    float* out = (float*)d_out;            // [0,480) kp_pos, [480,...) kp_feat
    float* ws  = (float*)d_ws;
    float* C_T = ws + WS_CT;
    float* S   = ws + WS_S;
    float* dk  = ws + WS_DK;
    int*   idx = (int*)(ws + WS_IDX);

    prep_kernel <<<dim3(BATCH * NCOL), dim3(256), 0, stream>>>(h0_kp, W_src, C_T, S);
    score_kernel<<<dim3(BATCH * 32),  dim3(256), 0, stream>>>(h_rec, x_rec, C_T, S);
    kppos_kernel<<<dim3(2),           dim3(256), 0, stream>>>(S, out);
    topk_kernel <<<dim3(BATCH * NKP), dim3(256), 0, stream>>>(x_rec, out, dk, idx);
    mlp_kernel  <<<dim3(BATCH * NKP), dim3(256), 0, stream>>>(h_rec, W_mlp, b_mlp,
                                                              ln_g, ln_b, dk, idx,
                                                              out + BATCH * NKP * 3);
}